// BoundaryAwareSegmentor_34488587387304
// MI455X (gfx1250) — compile-verified
//
#include <hip/hip_runtime.h>
#include <hip/hip_bf16.h>
#include <math.h>

// CDNA5 / gfx1250: wave32, WMMA 16x16 shapes.
typedef __attribute__((ext_vector_type(2))) float v2f;
typedef __attribute__((ext_vector_type(8))) float v8f;

#define TILE 16
#define TILES_PER_CHUNK 4
#define CHUNK_COLS (TILE * TILES_PER_CHUNK)      // 64 columns per barrier phase
#define ROWS_PER_WAVE 32
#define WAVES_PER_BLOCK 4
#define ROWS_PER_BLOCK (ROWS_PER_WAVE * WAVES_PER_BLOCK) // 128
// gram row stride: 67 = odd (scan lanes differ by 67%64=3, gcd(3,64)=1 -> conflict
// free) and 67%8==3 so lo/hi half-wave store rows differ by 8*67%64=24 -> the two
// 16-bank ranges are disjoint -> conflict-free stores.
#define GROW 67

// ---------------- kernel 1: squared norms ----------------
__global__ __launch_bounds__(256) void d2_kernel(const float* __restrict__ coord,
                                                 float* __restrict__ d2, int N) {
  int i = blockIdx.x * blockDim.x + threadIdx.x;
  if (i < N) {
    float x = coord[3 * i + 0];
    float y = coord[3 * i + 1];
    float z = coord[3 * i + 2];
    d2[i] = x * x + y * y + z * z;
  }
}

// ------- kernel 2: WMMA Gram tiles + streaming top-16 + boundary flag -------
// Software-pipelined: one barrier per 64-column chunk, double-buffered LDS,
// next chunk's global loads issued under the current chunk's scan.
__global__ __launch_bounds__(128) void knn_boundary_kernel(
    const float* __restrict__ coord, const int* __restrict__ seg,
    const float* __restrict__ d2, int* __restrict__ boundary, int N) {
  __shared__ float lds_gram[WAVES_PER_BLOCK * 2 * ROWS_PER_WAVE * GROW];
  __shared__ int lds_lab[WAVES_PER_BLOCK * 2 * CHUNK_COLS];

  const int lane = threadIdx.x & 31; // wave32
  const int wid = threadIdx.x >> 5;
  const int laneHalf = lane & 15;
  const bool hi = lane >= 16;

  const int i0w = blockIdx.x * ROWS_PER_BLOCK + wid * ROWS_PER_WAVE;
  float* gram0 = &lds_gram[wid * 2 * ROWS_PER_WAVE * GROW];
  float* gramB[2] = {gram0, gram0 + ROWS_PER_WAVE * GROW};
  int* lab0 = &lds_lab[wid * 2 * CHUNK_COLS];
  int* labB[2] = {lab0, lab0 + CHUNK_COLS};

  // A operands are loop-invariant (the wave's 32 rows), 16x4 f32 each:
  // lanes 0-15: VGPR0=K0(x), VGPR1=K1(y); lanes 16-31: VGPR0=K2(z), VGPR1=K3(0)
  const int rA0 = i0w + laneHalf;
  const int rA1 = rA0 + 16;
  {
    // branchless build
    const float x0 = coord[3 * rA0], y0 = coord[3 * rA0 + 1], z0 = coord[3 * rA0 + 2];
    const float x1 = coord[3 * rA1], y1 = coord[3 * rA1 + 1], z1 = coord[3 * rA1 + 2];
    (void)x0; (void)x1; // used below via selects
  }
  v2f a0, a1;
  {
    const float x0 = coord[3 * rA0], y0 = coord[3 * rA0 + 1], z0 = coord[3 * rA0 + 2];
    const float x1 = coord[3 * rA1], y1 = coord[3 * rA1 + 1], z1 = coord[3 * rA1 + 2];
    a0[0] = hi ? z0 : x0; a0[1] = hi ? 0.0f : y0;
    a1[0] = hi ? z1 : x1; a1[1] = hi ? 0.0f : y1;
  }

  const int myRow = i0w + lane; // lane owns one row of the 32-row strip
  const float d2i = d2[myRow];
  const int mySeg = seg[myRow];

  // register-resident sorted (ascending) top-16 of 64-bit keys:
  // key = orderable(dist) << 32 | (u16)label.   init = (+inf, label -1)
  unsigned long long k[16];
#pragma unroll
  for (int t = 0; t < 16; ++t) k[t] = 0xFF80000000000000ull | 0xFFFFull;
  float thresh = INFINITY; // == unpack(k[15] >> 32)

  // prefetched per-chunk column data (4 tiles)
  float pcx[TILES_PER_CHUNK], pcy[TILES_PER_CHUNK], pcz[TILES_PER_CHUNK],
      pd2[TILES_PER_CHUNK];
  int pseg[TILES_PER_CHUNK];

  auto do_loads = [&](int j0) {
#pragma unroll
    for (int t = 0; t < TILES_PER_CHUNK; ++t) {
      const int cB = j0 + t * TILE + laneHalf;
      pcx[t] = coord[3 * cB + 0];
      pcy[t] = coord[3 * cB + 1];
      pcz[t] = coord[3 * cB + 2];
      pd2[t] = d2[cB];
      pseg[t] = seg[cB];
    }
  };

  auto do_math = [&](float* gbuf, int* lbuf) {
#pragma unroll
    for (int t = 0; t < TILES_PER_CHUNK; ++t) {
      v2f b;
      b[0] = hi ? pcz[t] : pcx[t];
      b[1] = hi ? 0.0f : pcy[t];
      v8f c0 = {};
      v8f c1 = {};
      c0 = __builtin_amdgcn_wmma_f32_16x16x4_f32(false, a0, false, b, (short)0, c0, false, false);
      c1 = __builtin_amdgcn_wmma_f32_16x16x4_f32(false, a1, false, b, (short)0, c1, false, false);
      const int colBase = t * TILE + laneHalf;
      // single base + compile-time row offsets -> ds_store immediates
      float* srow = gbuf + colBase + (hi ? 8 * GROW : 0);
#pragma unroll
      for (int r = 0; r < 8; ++r) {
        srow[r * GROW] = fmaf(-2.0f, c0[r], pd2[t]);            // rows r / r+8
        srow[(r + 16) * GROW] = fmaf(-2.0f, c1[r], pd2[t]);     // rows r+16 / r+24
      }
      lbuf[colBase] = pseg[t]; // lanes l and l+16 write same value: benign
    }
  };

  // prologue: fill buffer 0
  do_loads(0);
  do_math(gramB[0], labB[0]);

  const int nChunks = N / CHUNK_COLS;
  for (int chunk = 0; chunk < nChunks; ++chunk) {
    __syncthreads(); // buf[cur] produced; buf[cur^1] free to overwrite
    const int cur = chunk & 1;
    const int j0 = chunk * CHUNK_COLS;
    const bool more = (chunk + 1 < nChunks);

    if (more) do_loads(j0 + CHUNK_COLS); // in flight during the scan below

    // self-exclusion: poison own diagonal entry (owner-lane private)
    const int dcol = myRow - j0;
    if (dcol >= 0 && dcol < CHUNK_COLS)
      gramB[cur][lane * GROW + dcol] = INFINITY;

    // scan: 64 candidates, 1 ds_load + 1 add + 1 cmp each
    const float* grow_p = gramB[cur] + lane * GROW;
    const int* labc = labB[cur];
#pragma unroll 4
    for (int n = 0; n < CHUNK_COLS; ++n) {
      const float dist = d2i + grow_p[n];
      if (dist < thresh) { // strict < implies key < k[15]
        unsigned u = __float_as_uint(dist);
        u ^= ((unsigned)((int)u >> 31)) | 0x80000000u; // total order on floats
        k[15] = ((unsigned long long)u << 32) |
                (unsigned long long)(unsigned short)labc[n];
#pragma unroll
        for (int t = 15; t >= 1; --t) { // bubble new key into place
          const unsigned long long x = k[t - 1], y = k[t];
          const bool sw = y < x;
          k[t - 1] = sw ? y : x;
          k[t] = sw ? x : y;
        }
        unsigned v = (unsigned)(k[15] >> 32); // refresh float threshold
        v = (v & 0x80000000u) ? (v ^ 0x80000000u) : ~v;
        thresh = __uint_as_float(v);
      }
    }

    if (more) do_math(gramB[cur ^ 1], labB[cur ^ 1]); // WMMA + stores for next
  }

  bool bnd = false;
#pragma unroll
  for (int t = 0; t < 16; ++t) {
    const int lab = (int)(short)(unsigned short)(k[t] & 0xFFFFull);
    bnd = bnd || ((lab != -1) && (lab != mySeg));
  }
  boundary[myRow] = ((mySeg != -1) && bnd) ? 1 : 0;
}

// -------- kernel 3: per-point CE + deterministic block partial sums --------
__global__ __launch_bounds__(256) void ce_partial_kernel(
    const float* __restrict__ logits, const int* __restrict__ seg,
    const int* __restrict__ boundary, float* __restrict__ partials, int N, int C) {
  __shared__ float sm[256];
  __shared__ float sc[256];
  __shared__ float sb[256];
  __shared__ float sq[256];

  const int tid = threadIdx.x;
  const int i = blockIdx.x * 256 + tid;
  float lsum = 0.f, lcnt = 0.f, bsum = 0.f, bcnt = 0.f;
  if (i < N) {
    const int t = seg[i];
    const bool valid = (t != -1);
    int tc = t < 0 ? 0 : (t > C - 1 ? C - 1 : t);
    const float* row = logits + (long long)i * C;
    float m = row[0];
    for (int c = 1; c < C; ++c) m = fmaxf(m, row[c]);
    float s = 0.f;
    for (int c = 0; c < C; ++c) s += expf(row[c] - m);
    const float lp = row[tc] - m - logf(s);
    if (valid) {
      lsum = lp; lcnt = 1.f;
      if (boundary[i]) { bsum = lp; bcnt = 1.f; }
    }
  }
  sm[tid] = lsum; sc[tid] = lcnt; sb[tid] = bsum; sq[tid] = bcnt;
  __syncthreads();
  for (int s = 128; s > 0; s >>= 1) {
    if (tid < s) {
      sm[tid] += sm[tid + s];
      sc[tid] += sc[tid + s];
      sb[tid] += sb[tid + s];
      sq[tid] += sq[tid + s];
    }
    __syncthreads();
  }
  if (tid == 0) {
    partials[4 * blockIdx.x + 0] = sm[0];
    partials[4 * blockIdx.x + 1] = sc[0];
    partials[4 * blockIdx.x + 2] = sb[0];
    partials[4 * blockIdx.x + 3] = sq[0];
  }
}

// ---------------- kernel 4: finalize scalar loss ----------------
__global__ void finalize_kernel(const float* __restrict__ partials,
                                float* __restrict__ out, int nb) {
  if (threadIdx.x == 0 && blockIdx.x == 0) {
    float sm = 0.f, sc = 0.f, sb = 0.f, sq = 0.f;
    for (int b = 0; b < nb; ++b) {
      sm += partials[4 * b + 0];
      sc += partials[4 * b + 1];
      sb += partials[4 * b + 2];
      sq += partials[4 * b + 3];
    }
    const float main_l = sc > 0.f ? -sm / sc : 0.f;
    const float bnd_l = sq > 0.f ? -sb / sq : 0.f;
    out[0] = main_l + 1.0f * bnd_l; // BOUNDARY_W = 1.0
  }
}

extern "C" void kernel_launch(void* const* d_in, const int* in_sizes, int n_in,
                              void* d_out, int out_size, void* d_ws, size_t ws_size,
                              hipStream_t stream) {
  const float* coord = (const float*)d_in[0];
  const float* seg_logits = (const float*)d_in[1];
  const int* segment = (const int*)d_in[2];
  // d_in[3] = offset scalar (== N), unused

  const int N = in_sizes[2];
  const int C = in_sizes[1] / N;

  char* ws = (char*)d_ws;
  float* d2 = (float*)ws;                              // N floats
  int* boundary = (int*)(ws + (size_t)N * 4);          // N ints
  float* partials = (float*)(ws + (size_t)2 * N * 4);  // 4 * nb floats

  float* out = (float*)d_out;

  const int nb = (N + 255) / 256;
  d2_kernel<<<nb, 256, 0, stream>>>(coord, d2, N);

  const int knnBlocks = N / ROWS_PER_BLOCK; // N = 16384 -> 128 blocks, 512 waves
  knn_boundary_kernel<<<knnBlocks, 128, 0, stream>>>(coord, segment, d2, boundary, N);

  ce_partial_kernel<<<nb, 256, 0, stream>>>(seg_logits, segment, boundary, partials, N, C);

  finalize_kernel<<<1, 1, 0, stream>>>(partials, out, nb);
}